// MemoryGCNConv_3_8169027797781
// MI455X (gfx1250) — compile-verified
//
#include <hip/hip_runtime.h>
#include <hip/hip_bf16.h>

#define N_NODES 50000
#define N_EDGES 800000
#define D 128
#define STRIPS (N_NODES / 16)   // 3125, exact

typedef __attribute__((ext_vector_type(2))) float v2f;
typedef __attribute__((ext_vector_type(8))) float v8f;
typedef unsigned int v4u __attribute__((ext_vector_type(4)));
typedef int v8i __attribute__((ext_vector_type(8)));
typedef int v4i __attribute__((ext_vector_type(4)));

// ---------------------------------------------------------------------------
// Workspace layout (floats):
//   deg/dinv : N_NODES                    (reused in place)
//   h        : N_NODES*D   (25.6 MB)
//   agg      : N_NODES*D   (25.6 MB)
//   colsum   : D   (becomes mu)
//   colsq    : D   (becomes gamma*rsqrt(var+eps))
// total ~51.4 MB
// ---------------------------------------------------------------------------

__global__ void __launch_bounds__(256)
init_kernel(float* __restrict__ deg, float* __restrict__ agg,
            float* __restrict__ colstats /* 2*D contiguous */) {
    size_t stride = (size_t)gridDim.x * blockDim.x;
    size_t total = (size_t)N_NODES * D;
    for (size_t j = (size_t)blockIdx.x * blockDim.x + threadIdx.x; j < total; j += stride) {
        agg[j] = 0.0f;
        if (j < N_NODES) deg[j] = 1.0f;       // self-loop contributes 1 to degree
        if (j < 2 * D)   colstats[j] = 0.0f;
    }
}

__global__ void __launch_bounds__(256)
degree_kernel(const int* __restrict__ dst, float* __restrict__ deg) {
    int e = blockIdx.x * blockDim.x + threadIdx.x;
    if (e < N_EDGES) atomicAdd(&deg[dst[e]], 1.0f);
}

__global__ void __launch_bounds__(256)
dinv_kernel(float* __restrict__ deg) {
    int i = blockIdx.x * blockDim.x + threadIdx.x;
    if (i < N_NODES) {
        float d = deg[i];
        deg[i] = d > 0.0f ? rsqrtf(d) : 0.0f;
    }
}

// Tensor Data Mover: DMA one 128x128 f32 matrix (64 KB) from global into LDS.
// D# per CDNA5 ISA ch.8: group0 = {count, lds_addr, global_addr, type=2},
// group1 = {data_size=4B, tensor_dim0/1=128, tile_dim0/1=128, dim0_stride=128},
// groups 2-3 zero (2D tensor). Tracked with TENSORcnt.
__device__ __forceinline__ void tdm_load_w(const float* gptr, unsigned lds_byte_off) {
    unsigned long long ga = (unsigned long long)(uintptr_t)gptr;
    v4u g0;
    g0.x = 0x1u;                                                // count=1, user desc
    g0.y = lds_byte_off;                                        // lds_addr (bytes)
    g0.z = (unsigned)(ga & 0xFFFFFFFFu);                        // global_addr[31:0]
    g0.w = (unsigned)((ga >> 32) & 0x01FFFFFFu) | 0x80000000u;  // addr[56:32] | type=2
    v8i g1;
    g1[0] = 0x00020000;            // data_size = 2 (4 bytes); no pad/iterate/multicast
    g1[1] = (int)(128u << 16);     // tensor_dim0 = 128 (low 16 bits)
    g1[2] = (int)(128u << 16);     // tensor_dim0 hi = 0 | tensor_dim1 = 128 (low 16)
    g1[3] = (int)(128u << 16);     // tensor_dim1 hi = 0 | tile_dim0 = 128
    g1[4] = 128;                   // tile_dim1 = 128, tile_dim2 = 0
    g1[5] = 128;                   // tensor_dim0_stride = 128 (elements)
    g1[6] = 0;                     // dim0_stride hi | tensor_dim1_stride lo
    g1[7] = 0;                     // tensor_dim1_stride hi
    v4i z4 = {0, 0, 0, 0};
    v8i z8 = {0, 0, 0, 0, 0, 0, 0, 0};
    __builtin_amdgcn_tensor_load_to_lds(g0, g1, z4, z4, z8, 0);
}

// Fused: h = x @ W_lin ; rv = leaky_relu(x @ W_mem, 0.01)
//        agg[row]            += dinv[row]^2 * h[row]        (self-loop term)
//        agg[recip[row]]     += rv[row]                     (memory scatter)
// One wave per 16-row strip; 8 column tiles of 16 -> full 16x128 strip in regs.
__global__ void __launch_bounds__(256)
gemm_fused_kernel(const float* __restrict__ x,
                  const float* __restrict__ Wlin,
                  const float* __restrict__ Wmem,
                  const float* __restrict__ dinv,
                  const int*   __restrict__ recip,
                  float* __restrict__ h,
                  float* __restrict__ agg) {
    extern __shared__ float lds[];   // [2 * D * D] : W_lin then W_mem, row-major [k][n]

    // TDM staging of both weight matrices into LDS (128 KB of 320 KB/WGP).
    // One wave issues both DMAs (EXEC is ignored by tensor ops), waits
    // TENSORcnt==0, then the whole block synchronizes.
    if (threadIdx.x == 0) {
        unsigned base = (unsigned)(uintptr_t)lds;   // LDS aperture keeps offset in [31:0]
        tdm_load_w(Wlin, base);
        tdm_load_w(Wmem, base + (unsigned)(D * D * sizeof(float)));
        __builtin_amdgcn_s_wait_tensorcnt(0);
    }
    __syncthreads();

    const int wave  = threadIdx.x >> 5;
    const int lane  = threadIdx.x & 31;
    const int strip = blockIdx.x * 8 + wave;
    if (strip >= STRIPS) return;

    const int row0 = strip * 16;
    const int half = lane >> 4;      // 0: lanes 0-15, 1: lanes 16-31
    const int l16  = lane & 15;

    // A fragment base: row = row0 + l16, K pair offset 2*half (ISA 16x4 f32 layout)
    const float* xrow = x + (size_t)(row0 + l16) * D + 2 * half;

    v8f acc_l[8], acc_m[8];
#pragma unroll
    for (int t = 0; t < 8; ++t) {
        acc_l[t] = (v8f){0.f,0.f,0.f,0.f,0.f,0.f,0.f,0.f};
        acc_m[t] = (v8f){0.f,0.f,0.f,0.f,0.f,0.f,0.f,0.f};
    }

    for (int k0 = 0; k0 < D; k0 += 4) {
        v2f a;
        a.x = xrow[k0];
        a.y = xrow[k0 + 1];
        const int kb = k0 + 2 * half;
#pragma unroll
        for (int t = 0; t < 8; ++t) {
            const int n = t * 16 + l16;
            v2f bl, bm;
            bl.x = lds[kb * D + n];
            bl.y = lds[(kb + 1) * D + n];
            bm.x = lds[D * D + kb * D + n];
            bm.y = lds[D * D + (kb + 1) * D + n];
            acc_l[t] = __builtin_amdgcn_wmma_f32_16x16x4_f32(
                false, a, false, bl, (short)0, acc_l[t], false, false);
            acc_m[t] = __builtin_amdgcn_wmma_f32_16x16x4_f32(
                false, a, false, bm, (short)0, acc_m[t], false, false);
        }
    }

    // D layout: vgpr v, lane -> row = row0 + v + 8*half, col = t*16 + l16
#pragma unroll
    for (int v = 0; v < 8; ++v) {
        const int r = row0 + v + 8 * half;
        const float di = dinv[r];
        const float wself = di * di;
        const int rc = recip[r];
#pragma unroll
        for (int t = 0; t < 8; ++t) {
            const int c = t * 16 + l16;
            float hv = acc_l[t][v];
            h[(size_t)r * D + c] = hv;
            atomicAdd(&agg[(size_t)r * D + c], wself * hv);
            float mv = acc_m[t][v];
            mv = mv > 0.0f ? mv : 0.01f * mv;
            atomicAdd(&agg[(size_t)rc * D + c], mv);
        }
    }
}

// One wave per edge: agg[dst] += dinv[src]*dinv[dst] * h[src]
__global__ void __launch_bounds__(256)
edge_kernel(const int* __restrict__ src, const int* __restrict__ dst,
            const float* __restrict__ dinv, const float* __restrict__ h,
            float* __restrict__ agg) {
    const int wave = threadIdx.x >> 5;
    const int lane = threadIdx.x & 31;
    const int e = blockIdx.x * 8 + wave;
    if (e >= N_EDGES) return;
    const int s = src[e];
    const int d = dst[e];
    const float w = dinv[s] * dinv[d];
    float4 v = ((const float4*)(h + (size_t)s * D))[lane];   // L2-resident gather
    float* ad = agg + (size_t)d * D + lane * 4;
    atomicAdd(ad + 0, w * v.x);
    atomicAdd(ad + 1, w * v.y);
    atomicAdd(ad + 2, w * v.z);
    atomicAdd(ad + 3, w * v.w);
}

// Column sums / sums of squares over 50000 rows.
__global__ void __launch_bounds__(128)
stats_kernel(const float* __restrict__ agg, float* __restrict__ colsum,
             float* __restrict__ colsq) {
    const int c = threadIdx.x;
    float s = 0.0f, s2 = 0.0f;
    for (int r = blockIdx.x; r < N_NODES; r += gridDim.x) {
        float v = agg[(size_t)r * D + c];
        s  += v;
        s2 += v * v;
    }
    atomicAdd(&colsum[c], s);
    atomicAdd(&colsq[c], s2);
}

// colsum -> mu ; colsq -> gamma * rsqrt(var + 1e-5)
__global__ void
statsfin_kernel(const float* __restrict__ gamma, float* __restrict__ colsum,
                float* __restrict__ colsq) {
    const int c = threadIdx.x;
    if (c < D) {
        const float inv_n = 1.0f / (float)N_NODES;
        float mu  = colsum[c] * inv_n;
        float var = colsq[c] * inv_n - mu * mu;
        colsum[c] = mu;
        colsq[c]  = gamma[c] * rsqrtf(var + 1e-5f);
    }
}

// BN affine -> ReLU -> row L2 normalize. One wave per row, 4 cols/lane.
__global__ void __launch_bounds__(256)
finalize_kernel(const float* __restrict__ agg, const float* __restrict__ mu,
                const float* __restrict__ gscale, const float* __restrict__ beta,
                float* __restrict__ out) {
    const int wave = threadIdx.x >> 5;
    const int lane = threadIdx.x & 31;
    const int r = blockIdx.x * 8 + wave;
    if (r >= N_NODES) return;
    const int c0 = lane * 4;
    float4 v = *(const float4*)(agg + (size_t)r * D + c0);
    float o[4];
    const float vin[4] = {v.x, v.y, v.z, v.w};
    float ss = 0.0f;
#pragma unroll
    for (int j = 0; j < 4; ++j) {
        const int c = c0 + j;
        float t = (vin[j] - mu[c]) * gscale[c] + beta[c];
        t = fmaxf(t, 0.0f);
        o[j] = t;
        ss += t * t;
    }
    // wave32 butterfly reduction
#pragma unroll
    for (int off = 16; off > 0; off >>= 1) ss += __shfl_xor(ss, off);
    const float inv = 1.0f / fmaxf(sqrtf(ss), 1e-12f);
    float4 ov = {o[0] * inv, o[1] * inv, o[2] * inv, o[3] * inv};
    *(float4*)(out + (size_t)r * D + c0) = ov;
}

extern "C" void kernel_launch(void* const* d_in, const int* in_sizes, int n_in,
                              void* d_out, int out_size, void* d_ws, size_t ws_size,
                              hipStream_t stream) {
    const float* x     = (const float*)d_in[0];
    const float* Wlin  = (const float*)d_in[1];
    const float* Wmem  = (const float*)d_in[2];
    const float* gamma = (const float*)d_in[3];
    const float* beta  = (const float*)d_in[4];
    const int*   edge  = (const int*)d_in[5];
    const int*   src   = edge;             // edge_index[0]
    const int*   dst   = edge + N_EDGES;   // edge_index[1]
    const int*   recip = (const int*)d_in[6];
    float*       out   = (float*)d_out;

    float* deg    = (float*)d_ws;                       // N_NODES (becomes dinv)
    float* h      = deg + N_NODES;                      // N*D
    float* agg    = h + (size_t)N_NODES * D;            // N*D
    float* colsum = agg + (size_t)N_NODES * D;          // D
    float* colsq  = colsum + D;                         // D

    init_kernel<<<2048, 256, 0, stream>>>(deg, agg, colsum);
    degree_kernel<<<(N_EDGES + 255) / 256, 256, 0, stream>>>(dst, deg);
    dinv_kernel<<<(N_NODES + 255) / 256, 256, 0, stream>>>(deg);
    gemm_fused_kernel<<<(STRIPS + 7) / 8, 256, 2 * D * D * sizeof(float), stream>>>(
        x, Wlin, Wmem, deg, recip, h, agg);
    edge_kernel<<<(N_EDGES + 7) / 8, 256, 0, stream>>>(src, dst, deg, h, agg);
    stats_kernel<<<1024, 128, 0, stream>>>(agg, colsum, colsq);
    statsfin_kernel<<<1, 128, 0, stream>>>(gamma, colsum, colsq);
    finalize_kernel<<<(N_NODES + 7) / 8, 256, 0, stream>>>(agg, colsum, colsq, beta, out);
}